// VectorQuantizer_57157424775536
// MI455X (gfx1250) — compile-verified
//
#include <hip/hip_runtime.h>
#include <hip/hip_bf16.h>
#include <math.h>

typedef __attribute__((ext_vector_type(2))) float v2f;
typedef __attribute__((ext_vector_type(8))) float v8f;

#define N_ROWS 65536
#define DIM 64
#define KEMB 1024
#define ROWS_PER_BLOCK 128
#define NUM_BLOCKS (N_ROWS / ROWS_PER_BLOCK)   // 512

// ---------------- Kernel 0: ||e_k||^2 and histogram init ----------------
__global__ void vq_init(const float* __restrict__ emb,
                        float* __restrict__ enorm,
                        unsigned* __restrict__ hist) {
    int k = blockIdx.x * blockDim.x + threadIdx.x;   // 0..1023
    float s = 0.f;
    #pragma unroll 8
    for (int d = 0; d < DIM; ++d) {
        float v = emb[d * KEMB + k];
        s += v * v;
    }
    enorm[k] = s;
    hist[k] = 0u;
}

// ---------------- Kernel 1: distances via f32 WMMA + argmin ----------------
__global__ __launch_bounds__(256)
void vq_main(const float* __restrict__ x,        // [65536,64]
             const float* __restrict__ emb,      // [64,1024]
             const float* __restrict__ enorm,    // [1024]
             float* __restrict__ quant,          // [65536,64]
             float* __restrict__ enc,            // [65536,1024]
             float* __restrict__ eidx,           // [65536]
             float* __restrict__ dist,           // [65536,1024]
             float* __restrict__ bpart,          // [512]
             unsigned* __restrict__ hist)        // [1024]
{
    __shared__ float wavesum[8];

    const int lane = threadIdx.x & 31;
    const int wave = threadIdx.x >> 5;
    const int m    = lane & 15;    // row within A-tile / col within B,C tile
    const int hi   = lane >> 4;    // half-wave select (K sub-pair / M+8)
    const int rowbase = blockIdx.x * ROWS_PER_BLOCK + wave * 16;
    const int row     = rowbase + m;

    // ---- Load A: 16 chunks of 16x4 f32. Lane m holds K = 4c+2*hi, 4c+2*hi+1.
    v2f a[16];
    const float* xr = x + (size_t)row * DIM + hi * 2;
    #pragma unroll
    for (int c = 0; c < 16; ++c)
        a[c] = *reinterpret_cast<const v2f*>(xr + 4 * c);

    // ---- ||x_row||^2 : per-lane partial + pair combine across halves
    float xn = 0.f;
    #pragma unroll
    for (int c = 0; c < 16; ++c) xn += a[c].x * a[c].x + a[c].y * a[c].y;
    xn += __shfl_xor(xn, 16, 32);
    // xnorm of the row each C-slot refers to (slot i -> row i + 8*hi)
    float xnrow[8];
    #pragma unroll
    for (int i = 0; i < 8; ++i) xnrow[i] = __shfl(xn, i + 8 * hi, 32);

    float best[8]; int bidx[8];
    #pragma unroll
    for (int i = 0; i < 8; ++i) { best[i] = 3.4e38f; bidx[i] = 0; }

    // ---- Column tiles: 64 tiles of 16 codebook entries
    for (int t = 0; t < KEMB / 16; ++t) {
        const int col = t * 16 + m;
        v8f acc = {};
        #pragma unroll
        for (int c = 0; c < 16; ++c) {
            const int kb = 4 * c + 2 * hi;
            v2f b;
            b.x = emb[(size_t)kb * KEMB + col];
            b.y = emb[(size_t)(kb + 1) * KEMB + col];
            // D = A(16x4) * B(4x16) + C, f32 -> v_wmma_f32_16x16x4_f32
            acc = __builtin_amdgcn_wmma_f32_16x16x4_f32(
                false, a[c], false, b, (short)0, acc, false, false);
        }
        const float en = enorm[col];
        #pragma unroll
        for (int i = 0; i < 8; ++i) {
            const int mm = i + 8 * hi;                 // C layout row
            const float dv = xnrow[i] - 2.0f * acc[i] + en;
            const size_t off = (size_t)(rowbase + mm) * KEMB + col;
            dist[off] = dv;
            enc[off]  = 0.0f;                          // one-hot zero fill
            if (dv < best[i]) { best[i] = dv; bidx[i] = col; }
        }
    }

    // ---- Argmin reduction across the 16 lanes of each half-wave
    float wsum = 0.f;
    #pragma unroll
    for (int i = 0; i < 8; ++i) {
        float bv = best[i]; int bi = bidx[i];
        #pragma unroll
        for (int off = 8; off >= 1; off >>= 1) {
            float ov = __shfl_xor(bv, off, 32);
            int   oi = __shfl_xor(bi, off, 32);
            if (ov < bv || (ov == bv && oi < bi)) { bv = ov; bi = oi; }
        }
        const int r = rowbase + i + 8 * hi;
        // gather quantized row: lane m writes d = m + 16j (coalesced per j)
        #pragma unroll
        for (int j = 0; j < 4; ++j) {
            const int d = m + 16 * j;
            quant[(size_t)r * DIM + d] = emb[(size_t)d * KEMB + bi];
        }
        if (m == 0) {
            eidx[r] = (float)bi;
            enc[(size_t)r * KEMB + bi] = 1.0f;         // one-hot fixup
            atomicAdd(hist + bi, 1u);                  // integer: deterministic
            wsum += bv;                                // min dist == row sq-error
        }
    }
    // deterministic per-block loss partial
    wsum += __shfl_xor(wsum, 16, 32);
    if (lane == 0) wavesum[wave] = wsum;
    __syncthreads();
    if (threadIdx.x == 0) {
        float s = 0.f;
        for (int w = 0; w < 8; ++w) s += wavesum[w];
        bpart[blockIdx.x] = s;
    }
}

// ---------------- Kernel 2: loss + perplexity ----------------
__global__ __launch_bounds__(1024)
void vq_finalize(const unsigned* __restrict__ hist,
                 const float* __restrict__ bpart,
                 float* __restrict__ loss_out,
                 float* __restrict__ ppl_out) {
    __shared__ float red[1024];
    const int t = threadIdx.x;
    const float p = (float)hist[t] * (1.0f / (float)N_ROWS);
    red[t] = p * logf(p + 1e-10f);
    __syncthreads();
    for (int s = 512; s > 0; s >>= 1) {
        if (t < s) red[t] += red[t + s];
        __syncthreads();
    }
    if (t == 0) {
        *ppl_out = expf(-red[0]);
        float sum = 0.f;
        for (int i = 0; i < NUM_BLOCKS; ++i) sum += bpart[i];  // fixed order
        *loss_out = 1.25f * sum / ((float)N_ROWS * (float)DIM);
    }
}

extern "C" void kernel_launch(void* const* d_in, const int* in_sizes, int n_in,
                              void* d_out, int out_size, void* d_ws, size_t ws_size,
                              hipStream_t stream) {
    const float* x   = (const float*)d_in[0];   // (64,32,32,64) fp32
    const float* emb = (const float*)d_in[1];   // (64,1024) fp32

    float* out   = (float*)d_out;
    float* quant = out;                                   // 4194304
    float* loss  = out + (size_t)N_ROWS * DIM;            // +1
    float* ppl   = loss + 1;                              // +1
    float* enc   = ppl + 1;                               // 67108864
    float* eidx  = enc + (size_t)N_ROWS * KEMB;           // 65536
    float* dst   = eidx + N_ROWS;                         // 67108864

    float*    enorm = (float*)d_ws;                       // [1024]
    unsigned* hist  = (unsigned*)d_ws + KEMB;             // [1024]
    float*    bprt  = (float*)d_ws + 2 * KEMB;            // [512]

    vq_init<<<KEMB / 256, 256, 0, stream>>>(emb, enorm, hist);
    vq_main<<<NUM_BLOCKS, 256, 0, stream>>>(x, emb, enorm, quant, enc, eidx,
                                            dst, bprt, hist);
    vq_finalize<<<1, 1024, 0, stream>>>(hist, bprt, loss, ppl);
}